// ChebClassifier_51531017617996
// MI455X (gfx1250) — compile-verified
//
#include <hip/hip_runtime.h>

// ---------------------------------------------------------------------------
// ChebNet classifier pipeline for gfx1250 (MI455X).
// Dense T_k @ W[k] GEMMs use V_WMMA_F32_16X16X4_F32 (fp32 matrix pipe).
// Sparse propagation / pooling use global float atomics.
// ---------------------------------------------------------------------------

typedef __attribute__((ext_vector_type(2))) float v2f;
typedef __attribute__((ext_vector_type(8))) float v8f;

enum {
  CN0 = 100000, CN1 = 25000, CN2 = 6250,
  CE0 = 600000, CE1 = 150000, CE2 = 37500,
  KCH = 6,
  CN1R = 25008,   // 25000 rounded up to 16
  CN2R = 6256     // 6250 rounded up to 16
};

// ----------------------------- elementwise helpers -------------------------

__global__ void k_zero(float* __restrict__ p, int n) {
  int i = blockIdx.x * blockDim.x + threadIdx.x;
  if (i < n) p[i] = 0.0f;
}

// x [N0,3] -> x4 [N0,4] zero-padded
__global__ void k_pad_x(const float* __restrict__ x, float* __restrict__ x4, int n) {
  int i = blockIdx.x * blockDim.x + threadIdx.x;
  if (i < n) {
    x4[i * 4 + 0] = x[i * 3 + 0];
    x4[i * 4 + 1] = x[i * 3 + 1];
    x4[i * 4 + 2] = x[i * 3 + 2];
    x4[i * 4 + 3] = 0.0f;
  }
}

// W0 [K,3,128] -> W0p [K,4,128] zero row 3
__global__ void k_pad_w0(const float* __restrict__ w, float* __restrict__ wp) {
  int i = blockIdx.x * blockDim.x + threadIdx.x;
  if (i < KCH * 4 * 128) {
    int n = i & 127;
    int rk = (i >> 7) & 3;
    int k = i >> 9;
    wp[i] = (rk < 3) ? w[(k * 3 + rk) * 128 + n] : 0.0f;
  }
}

// deg[src[e]] += 1
__global__ void k_degree(const int* __restrict__ src, float* __restrict__ deg, int e) {
  int i = blockIdx.x * blockDim.x + threadIdx.x;
  if (i < e) atomicAdd(&deg[src[i]], 1.0f);
}

// w[e] = -(dinv[src]*dinv[dst]),  dinv(v)= deg>0 ? rsqrt(deg) : 0
__global__ void k_edgew(const int* __restrict__ src, const int* __restrict__ dst,
                        const float* __restrict__ deg, float* __restrict__ w, int e) {
  int i = blockIdx.x * blockDim.x + threadIdx.x;
  if (i < e) {
    float ds = deg[src[i]];
    float dd = deg[dst[i]];
    float a = ds > 0.0f ? rsqrtf(ds) : 0.0f;
    float b = dd > 0.0f ? rsqrtf(dd) : 0.0f;
    w[i] = -(a * b);
  }
}

// out[dst[e]*F+f] += w[e] * h[src[e]*F+f]   (thread = (edge, feature))
__global__ void k_prop(const int* __restrict__ src, const int* __restrict__ dst,
                       const float* __restrict__ w, const float* __restrict__ h,
                       float* __restrict__ out, int e, int F) {
  int i = blockIdx.x * blockDim.x + threadIdx.x;
  if (i < e * F) {
    int ed = i / F;
    int f = i - ed * F;
    float v = w[ed] * h[src[ed] * F + f];
    atomicAdd(&out[dst[ed] * F + f], v);
  }
}

// Tn = 2*Tn - Tp  (Chebyshev recurrence completion)
__global__ void k_cheb(float* __restrict__ Tn, const float* __restrict__ Tp, int n) {
  int i = blockIdx.x * blockDim.x + threadIdx.x;
  if (i < n) Tn[i] = 2.0f * Tn[i] - Tp[i];
}

// out[rows[e]*F+f] += vals[e] * h[cols[e]*F+f]
__global__ void k_pool(const int* __restrict__ rows, const int* __restrict__ cols,
                       const float* __restrict__ vals, const float* __restrict__ h,
                       float* __restrict__ out, int nnz, int F) {
  int i = blockIdx.x * blockDim.x + threadIdx.x;
  if (i < nnz * F) {
    int e = i / F;
    int f = i - e * F;
    float v = vals[e] * h[cols[e] * F + f];
    atomicAdd(&out[rows[e] * F + f], v);
  }
}

// --------------------------- WMMA fp32 GEMM tile ---------------------------
// C[16x16 tile] (+)= A[Mpad,Kdim] @ B[Kdim,Nout]  via V_WMMA_F32_16X16X4_F32.
// One wave (32 lanes) per tile. Mpad, Nout multiples of 16; Kdim multiple of 4.
// accumulate==0: init acc with bias[col].  relu!=0: clamp at store.
__global__ void k_gemm_wmma(const float* __restrict__ A, const float* __restrict__ B,
                            const float* __restrict__ bias, float* __restrict__ C,
                            int Kdim, int Nout, int accumulate, int relu) {
  int lane = threadIdx.x;       // 0..31
  int half = lane >> 4;         // 0 or 1
  int r    = lane & 15;
  int row0 = blockIdx.x * 16;
  int col0 = blockIdx.y * 16;

  v8f acc;
  if (accumulate) {
#pragma unroll
    for (int v = 0; v < 8; ++v)
      acc[v] = C[(size_t)(row0 + 8 * half + v) * Nout + col0 + r];
  } else {
    float bv = bias[col0 + r];
#pragma unroll
    for (int v = 0; v < 8; ++v) acc[v] = bv;
  }

  for (int kk = 0; kk < Kdim; kk += 4) {
    // A 16x4: lane r holds row (row0+r); half selects K pair {0,1} vs {2,3}
    const float* arow = A + (size_t)(row0 + r) * Kdim + kk + 2 * half;
    v2f a;
    a.x = arow[0];
    a.y = arow[1];
    // B 4x16: lane r holds col (col0+r); half selects K pair
    const float* brow = B + (size_t)(kk + 2 * half) * Nout + col0 + r;
    v2f b;
    b.x = brow[0];
    b.y = brow[Nout];
    acc = __builtin_amdgcn_wmma_f32_16x16x4_f32(
        /*neg_a=*/false, a, /*neg_b=*/false, b,
        /*c_mod=*/(short)0, acc, /*reuse_a=*/false, /*reuse_b=*/false);
  }

#pragma unroll
  for (int v = 0; v < 8; ++v) {
    float val = acc[v];
    if (relu) val = fmaxf(val, 0.0f);
    C[(size_t)(row0 + 8 * half + v) * Nout + col0 + r] = val;
  }
}

// ------------------------------ final linear -------------------------------

__global__ void k_out_init(const float* __restrict__ linb, float* __restrict__ out) {
  int i = threadIdx.x;
  if (i < 10) out[i] = linb[i];
}

// Z[o] += sum_i linW[o,i]*h[i];  grid.y = 10 outputs, grid.x = chunks of 4096
__global__ void k_linear(const float* __restrict__ linW, const float* __restrict__ h,
                         float* __restrict__ out, int len) {
  __shared__ float s[256];
  int o = blockIdx.y;
  const float* wrow = linW + (size_t)o * len;
  int base = blockIdx.x * (256 * 16) + threadIdx.x;
  float sum = 0.0f;
#pragma unroll
  for (int j = 0; j < 16; ++j) {
    int idx = base + j * 256;
    if (idx < len) sum += wrow[idx] * h[idx];
  }
  s[threadIdx.x] = sum;
  __syncthreads();
  for (int stp = 128; stp > 0; stp >>= 1) {
    if (threadIdx.x < stp) s[threadIdx.x] += s[threadIdx.x + stp];
    __syncthreads();
  }
  if (threadIdx.x == 0) atomicAdd(&out[o], s[0]);
}

// ------------------------------- host driver -------------------------------

extern "C" void kernel_launch(void* const* d_in, const int* in_sizes, int n_in,
                              void* d_out, int out_size, void* d_ws, size_t ws_size,
                              hipStream_t stream) {
  const float* x    = (const float*)d_in[0];
  const int*   ei0  = (const int*)d_in[1];
  const int*   ei1  = (const int*)d_in[2];
  const int*   ei2  = (const int*)d_in[3];
  const float* W0   = (const float*)d_in[4];
  const float* b0   = (const float*)d_in[5];
  const float* W1   = (const float*)d_in[6];
  const float* b1   = (const float*)d_in[7];
  const float* W2   = (const float*)d_in[8];
  const float* b2   = (const float*)d_in[9];
  const int*   D0r  = (const int*)d_in[10];
  const int*   D0c  = (const int*)d_in[11];
  const float* D0v  = (const float*)d_in[12];
  const int*   D1r  = (const int*)d_in[13];
  const int*   D1c  = (const int*)d_in[14];
  const float* D1v  = (const float*)d_in[15];
  const float* linW = (const float*)d_in[16];
  const float* linb = (const float*)d_in[17];
  float* Z = (float*)d_out;
  (void)in_sizes; (void)n_in; (void)out_size; (void)ws_size;

  // workspace carve-up (floats, 16B aligned chunks)
  size_t off = 0;
  auto alloc = [&](size_t n) -> float* {
    float* p = (float*)d_ws + off;
    off += (n + 3) & ~(size_t)3;
    return p;
  };
  float* x4   = alloc((size_t)CN0 * 4);
  float* W0p  = alloc((size_t)KCH * 4 * 128);
  float* deg0 = alloc(CN0);
  float* w0   = alloc(CE0);
  float* T0b  = alloc((size_t)CN0 * 4);
  float* T0c  = alloc((size_t)CN0 * 4);
  float* out0 = alloc((size_t)CN0 * 128);
  float* h1   = alloc((size_t)CN1R * 128);
  float* deg1 = alloc(CN1);
  float* w1   = alloc(CE1);
  float* T1b  = alloc((size_t)CN1R * 128);
  float* T1c  = alloc((size_t)CN1R * 128);
  float* out1 = alloc((size_t)CN1R * 128);
  float* h2   = alloc((size_t)CN2R * 128);
  float* deg2 = alloc(CN2);
  float* w2   = alloc(CE2);
  float* T2b  = alloc((size_t)CN2R * 128);
  float* T2c  = alloc((size_t)CN2R * 128);
  float* out2 = alloc((size_t)CN2R * 256);

  auto zero = [&](float* p, int n) {
    k_zero<<<(n + 255) / 256, 256, 0, stream>>>(p, n);
  };

  // One ChebConv layer: A0 = Tx0 (padded, also used as rotation scratch).
  auto cheb_layer = [&](float* A0, float* Tb, float* Tc,
                        const int* src, const int* dst,
                        float* deg, float* wbuf, int nNodes, int Mpad, int E,
                        const float* Wmat, const float* bias, float* Cout,
                        int Fin, int Fout, int relu) {
    zero(deg, nNodes);
    k_degree<<<(E + 255) / 256, 256, 0, stream>>>(src, deg, E);
    k_edgew<<<(E + 255) / 256, 256, 0, stream>>>(src, dst, deg, wbuf, E);

    dim3 g(Mpad / 16, Fout / 16);
    // k = 0: out = Tx0 @ W[0] + bias
    k_gemm_wmma<<<g, 32, 0, stream>>>(A0, Wmat, bias, Cout, Fin, Fout, 0, 0);

    int tp = E * Fin;
    // k = 1: Tx1 = L_hat Tx0
    zero(Tb, Mpad * Fin);
    k_prop<<<(tp + 255) / 256, 256, 0, stream>>>(src, dst, wbuf, A0, Tb, E, Fin);
    k_gemm_wmma<<<g, 32, 0, stream>>>(Tb, Wmat + (size_t)1 * Fin * Fout, bias,
                                      Cout, Fin, Fout, 1, 0);

    float* prev = A0;
    float* cur = Tb;
    float* nxt = Tc;
    for (int k = 2; k < KCH; ++k) {
      zero(nxt, Mpad * Fin);
      k_prop<<<(tp + 255) / 256, 256, 0, stream>>>(src, dst, wbuf, cur, nxt, E, Fin);
      k_cheb<<<(Mpad * Fin + 255) / 256, 256, 0, stream>>>(nxt, prev, Mpad * Fin);
      int do_relu = (relu && k == KCH - 1) ? 1 : 0;
      k_gemm_wmma<<<g, 32, 0, stream>>>(nxt, Wmat + (size_t)k * Fin * Fout, bias,
                                        Cout, Fin, Fout, 1, do_relu);
      float* t = prev; prev = cur; cur = nxt; nxt = t;
    }
  };

  // -------- layer 0: [N0,3] -> relu -> [N0,128] --------
  k_pad_x<<<(CN0 + 255) / 256, 256, 0, stream>>>(x, x4, CN0);
  k_pad_w0<<<(KCH * 4 * 128 + 255) / 256, 256, 0, stream>>>(W0, W0p);
  cheb_layer(x4, T0b, T0c, ei0, ei0 + CE0, deg0, w0, CN0, CN0, CE0,
             W0p, b0, out0, 4, 128, 1);

  // pool 0 -> h1 [N1R,128]
  zero(h1, CN1R * 128);
  k_pool<<<(100000 * 128 + 255) / 256, 256, 0, stream>>>(D0r, D0c, D0v, out0, h1,
                                                          100000, 128);

  // -------- layer 1: [N1,128] -> relu -> [N1,128] --------
  cheb_layer(h1, T1b, T1c, ei1, ei1 + CE1, deg1, w1, CN1, CN1R, CE1,
             W1, b1, out1, 128, 128, 1);

  // pool 1 -> h2 [N2R,128]
  zero(h2, CN2R * 128);
  k_pool<<<(25000 * 128 + 255) / 256, 256, 0, stream>>>(D1r, D1c, D1v, out1, h2,
                                                         25000, 128);

  // -------- layer 2: [N2,128] -> [N2,256] (no relu) --------
  cheb_layer(h2, T2b, T2c, ei2, ei2 + CE2, deg2, w2, CN2, CN2R, CE2,
             W2, b2, out2, 128, 256, 0);

  // -------- final linear: Z = linW @ out2.flat + linb --------
  k_out_init<<<1, 16, 0, stream>>>(linb, Z);
  int len = CN2 * 256;  // real rows are contiguous; padding rows sit after them
  dim3 lg((len + 4095) / 4096, 10);
  k_linear<<<lg, 256, 0, stream>>>(linW, out2, Z, len);
}